// GraphGoldNet_6262062317986
// MI455X (gfx1250) — compile-verified
//
#include <hip/hip_runtime.h>

typedef float v2f __attribute__((ext_vector_type(2)));
typedef float v8f __attribute__((ext_vector_type(8)));
typedef unsigned int v4u __attribute__((ext_vector_type(4)));
typedef int v8i __attribute__((ext_vector_type(8)));
typedef int v4i __attribute__((ext_vector_type(4)));

#define CDIV(a, b) (((a) + (b) - 1) / (b))

static constexpr int C = 128;
static constexpr float NEGBIG = -1e9f;

// ===================== elementwise utilities =====================
__global__ void fill_f32_kernel(float* __restrict__ p, float v, int n) {
  int i = blockIdx.x * blockDim.x + threadIdx.x;
  if (i < n) p[i] = v;
}
__global__ void fill_i32_kernel(int* __restrict__ p, int v, int n) {
  int i = blockIdx.x * blockDim.x + threadIdx.x;
  if (i < n) p[i] = v;
}
__global__ void copy_kernel(const float* __restrict__ a, float* __restrict__ o, int n) {
  int i = blockIdx.x * blockDim.x + threadIdx.x;
  if (i < n) o[i] = a[i];
}
__global__ void add_kernel(const float* __restrict__ a, const float* __restrict__ b,
                           float* __restrict__ o, int n) {
  int i = blockIdx.x * blockDim.x + threadIdx.x;
  if (i < n) o[i] = a[i] + b[i];
}
__global__ void relu_kernel(float* __restrict__ a, int n) {
  int i = blockIdx.x * blockDim.x + threadIdx.x;
  if (i < n) a[i] = fmaxf(a[i], 0.0f);
}

// ===================== WMMA GEMM =====================
// Out[n,128] = X[n,128] @ W[128,128], fp32 exact via V_WMMA_F32_16X16X4_F32.
// Block = 128 threads = 4 waves; each wave owns a 16-row x 128-col strip.
// W is staged into LDS by the Tensor Data Mover (TENSOR_LOAD_TO_LDS + TENSORcnt)
// when the TDM builtin is available; otherwise by a b128 copy loop.
__global__ __launch_bounds__(128) void gemm128_wmma_kernel(
    const float* __restrict__ X, const float* __restrict__ W,
    float* __restrict__ Out, int n) {
  __shared__ float lW[128 * 128];  // 64 KB (WGP has 320 KB)

#if __has_builtin(__builtin_amdgcn_tensor_load_to_lds) && \
    __has_builtin(__builtin_amdgcn_s_wait_tensorcnt)
  // TDM staging: one wave issues the DMA (TDM ignores EXEC; gate by wave to
  // avoid redundant transfers), waits on TENSORcnt, then the block barriers.
  if (threadIdx.x < 32) {
    // Flat shared pointers carry the LDS byte offset in addr[31:0].
    unsigned lds_off = (unsigned)(unsigned long long)(const void*)lW;
    unsigned long long ga = (unsigned long long)(const void*)W;
    // D# group 0: count=1 (valid), lds_addr, global_addr[56:0], type=2
    v4u g0;
    g0.x = 1u;
    g0.y = lds_off;
    g0.z = (unsigned)ga;
    g0.w = (unsigned)((ga >> 32) & 0x01FFFFFFu) | (2u << 30);
    // D# group 1: data_size=4B(code 2), tensor_dim0=128, tensor_dim1=128,
    //             tile_dim0=128, tile_dim1=128, tile_dim2=0, dim0_stride=128
    v8i g1;
    g1[0] = (int)(2u << 16);    // workgroup_mask=0, data_size=2 (4 bytes)
    g1[1] = (int)(128u << 16);  // [63:48] = tensor_dim0 lo16
    g1[2] = (int)(128u << 16);  // [79:64]=tensor_dim0 hi16(0), [95:80]=dim1 lo16
    g1[3] = (int)(128u << 16);  // [111:96]=tensor_dim1 hi16(0), [127:112]=tile_dim0
    g1[4] = 128;                // [143:128]=tile_dim1, [159:144]=tile_dim2=0
    g1[5] = 128;                // tensor_dim0_stride[31:0]
    g1[6] = 0;                  // stride0 hi16 / dim1_stride lo16
    g1[7] = 0;
    v4i z4 = (v4i)0;            // groups 2/3 unused (2D tensor)
    v8i z8 = (v8i)0;            // extra group (6-arg toolchain form)
    __builtin_amdgcn_tensor_load_to_lds(g0, g1, z4, z4, z8, 0);
    __builtin_amdgcn_s_wait_tensorcnt(0);
  }
  __syncthreads();
#else
  for (int i = threadIdx.x; i < 128 * 128 / 4; i += 128)
    ((float4*)lW)[i] = ((const float4*)W)[i];
  __syncthreads();
#endif

  const int lane = threadIdx.x & 31;
  const int wave = threadIdx.x >> 5;
  const int m = lane & 15;
  const int hi = lane >> 4;  // 0: K=k0,k0+1 ; 1: K=k0+2,k0+3
  const int rowBase = blockIdx.x * 64 + wave * 16;
  const int row = rowBase + m;
  const bool inb = row < n;
  const float* __restrict__ arow = X + (size_t)(inb ? row : 0) * C;
  __builtin_prefetch(arow, 0, 1);

  v8f acc[8];
#pragma unroll
  for (int t = 0; t < 8; ++t) acc[t] = (v8f)0.0f;

  for (int k0 = 0; k0 < 128; k0 += 4) {
    v2f a;
    a.x = arow[k0 + 2 * hi + 0];
    a.y = arow[k0 + 2 * hi + 1];
    if (!inb) { a.x = 0.0f; a.y = 0.0f; }
#pragma unroll
    for (int t = 0; t < 8; ++t) {
      v2f b;
      b.x = lW[(k0 + 2 * hi + 0) * 128 + t * 16 + m];
      b.y = lW[(k0 + 2 * hi + 1) * 128 + t * 16 + m];
      acc[t] = __builtin_amdgcn_wmma_f32_16x16x4_f32(
          false, a, false, b, (short)0, acc[t], false, false);
    }
  }
  // C/D layout: VGPR r -> row r (lanes0-15) / row r+8 (lanes16-31), col = tile*16 + (lane&15)
#pragma unroll
  for (int t = 0; t < 8; ++t) {
    const int col = t * 16 + m;
#pragma unroll
    for (int r = 0; r < 8; ++r) {
      const int orow = rowBase + r + 8 * hi;
      if (orow < n) Out[(size_t)orow * C + col] = acc[t][r];
    }
  }
}

// ===================== GCN =====================
__global__ void gcn_deg_kernel(const int* __restrict__ src, const int* __restrict__ dst,
                               const float* __restrict__ w, const int* __restrict__ mask,
                               float* __restrict__ deg, int E) {
  int e = blockIdx.x * blockDim.x + threadIdx.x;
  if (e >= E || !mask[e]) return;
  atomicAdd(&deg[dst[e]], w[e]);
}
__global__ void gcn_dinv_kernel(const float* __restrict__ deg, float* __restrict__ dinv, int n) {
  int i = blockIdx.x * blockDim.x + threadIdx.x;
  if (i < n) dinv[i] = rsqrtf(deg[i] + 2.0f);
}
// out[i][c] = b[c] + 2*dinv[i]^2 * h[i][c]
__global__ void gcn_init_kernel(const float* __restrict__ h, const float* __restrict__ bias,
                                const float* __restrict__ dinv, float* __restrict__ out, int n) {
  int idx = blockIdx.x * blockDim.x + threadIdx.x;
  if (idx >= n * C) return;
  int i = idx >> 7, c = idx & 127;
  float di = dinv[i];
  out[idx] = bias[c] + 2.0f * di * di * h[idx];
}
// wave per edge: out[dst] += norm * h[src]
__global__ __launch_bounds__(256) void gcn_scatter_kernel(
    const int* __restrict__ src, const int* __restrict__ dst,
    const float* __restrict__ w, const int* __restrict__ mask,
    const float* __restrict__ dinv, const float* __restrict__ h,
    float* __restrict__ out, int E) {
  int e = blockIdx.x * 8 + (threadIdx.x >> 5);
  int lane = threadIdx.x & 31;
  if (e >= E || !mask[e]) return;
  int s = src[e], d = dst[e];
  float nrm = dinv[s] * w[e] * dinv[d];
  float4 v = ((const float4*)(h + (size_t)s * C))[lane];
  float* od = out + (size_t)d * C + lane * 4;
  atomicAdd(od + 0, nrm * v.x);
  atomicAdd(od + 1, nrm * v.y);
  atomicAdd(od + 2, nrm * v.z);
  atomicAdd(od + 3, nrm * v.w);
}

// ===================== GAT =====================
__device__ __forceinline__ void atomicMaxF(float* addr, float v) {
  if (v >= 0.0f) atomicMax((int*)addr, __float_as_int(v));
  else           atomicMin((unsigned int*)addr, (unsigned int)__float_as_int(v));
}
__device__ __forceinline__ float leaky02(float v) { return v > 0.0f ? v : 0.2f * v; }

// per (node, head): al_s/al_d dot products over 32-wide head slices
__global__ void gat_alpha_kernel(const float* __restrict__ h, const float* __restrict__ asrc,
                                 const float* __restrict__ adst, float* __restrict__ al_s,
                                 float* __restrict__ al_d, int n) {
  int idx = blockIdx.x * blockDim.x + threadIdx.x;
  if (idx >= n * 4) return;
  int i = idx >> 2, hh = idx & 3;
  const float* hp = h + (size_t)i * C + hh * 32;
  const float* as = asrc + hh * 32;
  const float* ad = adst + hh * 32;
  float s = 0.0f, d = 0.0f;
#pragma unroll 8
  for (int c = 0; c < 32; ++c) { float v = hp[c]; s += v * as[c]; d += v * ad[c]; }
  al_s[idx] = s; al_d[idx] = d;
}
// entries m in [0, E+n): m<E -> edge, else self-loop m-E
__global__ void gat_max_kernel(const int* __restrict__ src, const int* __restrict__ dst,
                               const int* __restrict__ mask, const float* __restrict__ al_s,
                               const float* __restrict__ al_d, float* __restrict__ mx,
                               int E, int n) {
  int idx = blockIdx.x * blockDim.x + threadIdx.x;
  if (idx >= (E + n) * 4) return;
  int m = idx >> 2, hh = idx & 3;
  int s, d;
  if (m < E) { if (!mask[m]) return; s = src[m]; d = dst[m]; }
  else       { s = d = m - E; }
  atomicMaxF(&mx[d * 4 + hh], leaky02(al_s[s * 4 + hh] + al_d[d * 4 + hh]));
}
__global__ void gat_exp_kernel(const int* __restrict__ src, const int* __restrict__ dst,
                               const int* __restrict__ mask, const float* __restrict__ al_s,
                               const float* __restrict__ al_d, const float* __restrict__ mx,
                               float* __restrict__ e_val, float* __restrict__ denom,
                               int E, int n) {
  int idx = blockIdx.x * blockDim.x + threadIdx.x;
  if (idx >= (E + n) * 4) return;
  int m = idx >> 2, hh = idx & 3;
  int s, d; bool valid;
  if (m < E) { valid = mask[m] != 0; s = src[m]; d = dst[m]; }
  else       { valid = true; s = d = m - E; }
  float e = 0.0f;
  if (valid) {
    e = __expf(leaky02(al_s[s * 4 + hh] + al_d[d * 4 + hh]) - mx[d * 4 + hh]);
    atomicAdd(&denom[d * 4 + hh], e);
  }
  e_val[idx] = e;
}
__global__ void init_bias_kernel(const float* __restrict__ bias, float* __restrict__ out, int n) {
  int idx = blockIdx.x * blockDim.x + threadIdx.x;
  if (idx < n * C) out[idx] = bias[idx & 127];
}
// wave per entry: out[d] += att * h[s]
__global__ __launch_bounds__(256) void gat_scatter_kernel(
    const int* __restrict__ src, const int* __restrict__ dst, const int* __restrict__ mask,
    const float* __restrict__ e_val, const float* __restrict__ denom,
    const float* __restrict__ h, float* __restrict__ out, int E, int n) {
  int m = blockIdx.x * 8 + (threadIdx.x >> 5);
  int lane = threadIdx.x & 31;
  if (m >= E + n) return;
  int s, d;
  if (m < E) { if (!mask[m]) return; s = src[m]; d = dst[m]; }
  else       { s = d = m - E; }
#pragma unroll
  for (int hh = 0; hh < 4; ++hh) {
    float att = e_val[m * 4 + hh] / denom[d * 4 + hh];
    int c = hh * 32 + lane;
    atomicAdd(&out[(size_t)d * C + c], att * h[(size_t)s * C + c]);
  }
}

// ===================== TopK pooling =====================
__global__ void pw_invnorm_kernel(const float* __restrict__ pw, float* __restrict__ out) {
  __shared__ float s[128];
  float v = pw[threadIdx.x];
  s[threadIdx.x] = v * v;
  __syncthreads();
  for (int off = 64; off > 0; off >>= 1) {
    if (threadIdx.x < off) s[threadIdx.x] += s[threadIdx.x + off];
    __syncthreads();
  }
  if (threadIdx.x == 0) out[0] = rsqrtf(s[0]);
}
__global__ void score_kernel(const float* __restrict__ x, const float* __restrict__ pw,
                             const float* __restrict__ invn, float* __restrict__ score, int n) {
  int i = blockIdx.x * blockDim.x + threadIdx.x;
  if (i >= n) return;
  const float* xi = x + (size_t)i * C;
  float acc = 0.0f;
#pragma unroll 8
  for (int c = 0; c < C; ++c) acc += xi[c] * pw[c];
  score[i] = tanhf(acc * invn[0]);
}
// rank[i] = #{j: s_j > s_i  or (s_j == s_i and j < i)}  (stable descending order)
__global__ __launch_bounds__(256) void rank_kernel(const float* __restrict__ score,
                                                   int* __restrict__ rank, int n) {
  __shared__ float tile[256];
  int i = blockIdx.x * 256 + threadIdx.x;
  float si = (i < n) ? score[i] : 0.0f;
  int r = 0;
  for (int base = 0; base < n; base += 256) {
    int j = base + threadIdx.x;
    tile[threadIdx.x] = (j < n) ? score[j] : -1e30f;
    __syncthreads();
    int lim = min(256, n - base);
    for (int t = 0; t < lim; ++t) {
      float sj = tile[t];
      int jj = base + t;
      if (sj > si || (sj == si && jj < i)) r++;
    }
    __syncthreads();
  }
  if (i < n) rank[i] = r;
}
__global__ void build_perm_kernel(const int* __restrict__ rank, int* __restrict__ pos,
                                  int* __restrict__ perm, int n, int k) {
  int i = blockIdx.x * blockDim.x + threadIdx.x;
  if (i >= n) return;
  int r = rank[i];
  if (r < k) { pos[i] = r; perm[r] = i; } else pos[i] = -1;
}
__global__ void gather_scale_kernel(const float* __restrict__ x, const int* __restrict__ perm,
                                    const float* __restrict__ score, float* __restrict__ xn,
                                    int k) {
  int r = blockIdx.x, c = threadIdx.x;
  int i = perm[r];
  xn[(size_t)r * C + c] = x[(size_t)i * C + c] * score[i];
}
__global__ void edge_remap_kernel(const int* __restrict__ src, const int* __restrict__ dst,
                                  const int* __restrict__ mask, const int* __restrict__ pos,
                                  int* __restrict__ nsrc, int* __restrict__ ndst,
                                  int* __restrict__ nmask, int E) {
  int e = blockIdx.x * blockDim.x + threadIdx.x;
  if (e >= E) return;
  int ps = pos[src[e]], pd = pos[dst[e]];
  int m = (mask[e] && ps >= 0 && pd >= 0) ? 1 : 0;
  nsrc[e] = m ? ps : 0;
  ndst[e] = m ? pd : 0;
  nmask[e] = m;
}
// out[perm[r]] += xs[r]  (perm entries unique -> plain RMW per element is fine)
__global__ void unpool_scatter_kernel(const int* __restrict__ perm, const float* __restrict__ xs,
                                      float* __restrict__ out, int k) {
  int idx = blockIdx.x * blockDim.x + threadIdx.x;
  if (idx >= k * C) return;
  int r = idx >> 7, c = idx & 127;
  out[(size_t)perm[r] * C + c] += xs[idx];
}

// ===================== host-side composites =====================
static void run_gcn(hipStream_t st, const float* xin, const int* src, const int* dst,
                    const float* ew, const int* mask, const float* Wm, const float* bias,
                    float* out, int n, int E, float* deg, float* dinv, float* h, bool do_relu) {
  fill_f32_kernel<<<CDIV(n, 256), 256, 0, st>>>(deg, 0.0f, n);
  gcn_deg_kernel<<<CDIV(E, 256), 256, 0, st>>>(src, dst, ew, mask, deg, E);
  gcn_dinv_kernel<<<CDIV(n, 256), 256, 0, st>>>(deg, dinv, n);
  gemm128_wmma_kernel<<<CDIV(n, 64), 128, 0, st>>>(xin, Wm, h, n);
  gcn_init_kernel<<<CDIV(n * C, 256), 256, 0, st>>>(h, bias, dinv, out, n);
  gcn_scatter_kernel<<<CDIV(E, 8), 256, 0, st>>>(src, dst, ew, mask, dinv, h, out, E);
  if (do_relu) relu_kernel<<<CDIV(n * C, 256), 256, 0, st>>>(out, n * C);
}

static void run_gat(hipStream_t st, const float* xin, const int* src, const int* dst,
                    const int* mask, const float* Wm, const float* asrc, const float* adst,
                    const float* bias, float* out, int n, int E,
                    float* h, float* al_s, float* al_d, float* mx, float* denom, float* e_val) {
  gemm128_wmma_kernel<<<CDIV(n, 64), 128, 0, st>>>(xin, Wm, h, n);
  gat_alpha_kernel<<<CDIV(n * 4, 256), 256, 0, st>>>(h, asrc, adst, al_s, al_d, n);
  fill_f32_kernel<<<CDIV(n * 4, 256), 256, 0, st>>>(mx, NEGBIG, n * 4);
  fill_f32_kernel<<<CDIV(n * 4, 256), 256, 0, st>>>(denom, 0.0f, n * 4);
  int M4 = (E + n) * 4;
  gat_max_kernel<<<CDIV(M4, 256), 256, 0, st>>>(src, dst, mask, al_s, al_d, mx, E, n);
  gat_exp_kernel<<<CDIV(M4, 256), 256, 0, st>>>(src, dst, mask, al_s, al_d, mx, e_val, denom, E, n);
  init_bias_kernel<<<CDIV(n * C, 256), 256, 0, st>>>(bias, out, n);
  gat_scatter_kernel<<<CDIV(E + n, 8), 256, 0, st>>>(src, dst, mask, e_val, denom, h, out, E, n);
  relu_kernel<<<CDIV(n * C, 256), 256, 0, st>>>(out, n * C);
}

static void run_pool(hipStream_t st, const float* xin, int n, int k, const float* pw,
                     float* invn, float* score, int* rank, int* pos, int* perm, float* xout,
                     const int* src, const int* dst, const int* mask,
                     int* nsrc, int* ndst, int* nmask, int E) {
  pw_invnorm_kernel<<<1, 128, 0, st>>>(pw, invn);
  score_kernel<<<CDIV(n, 128), 128, 0, st>>>(xin, pw, invn, score, n);
  rank_kernel<<<CDIV(n, 256), 256, 0, st>>>(score, rank, n);
  build_perm_kernel<<<CDIV(n, 256), 256, 0, st>>>(rank, pos, perm, n, k);
  gather_scale_kernel<<<k, 128, 0, st>>>(xin, perm, score, xout, k);
  edge_remap_kernel<<<CDIV(E, 256), 256, 0, st>>>(src, dst, mask, pos, nsrc, ndst, nmask, E);
}

// ===================== entry point =====================
extern "C" void kernel_launch(void* const* d_in, const int* in_sizes, int n_in,
                              void* d_out, int out_size, void* d_ws, size_t ws_size,
                              hipStream_t stream) {
  const int n0 = 25000, E = 400000, k1 = 12500, k2 = 6250;
  const int n1 = k1, n2 = k2;

  const float* x    = (const float*)d_in[0];
  const int*   src0 = (const int*)d_in[1];
  const int*   dst0 = src0 + E;
  const float* ew   = (const float*)d_in[2];
  const float* dgW  = (const float*)d_in[3];   // [3][C][C]
  const float* dgb  = (const float*)d_in[4];   // [3][C]
  const float* gaW  = (const float*)d_in[5];   // [3][C][C]
  const float* gas  = (const float*)d_in[6];   // [3][128]
  const float* gad  = (const float*)d_in[7];
  const float* gab  = (const float*)d_in[8];
  const float* plw  = (const float*)d_in[9];   // [2][C]
  const float* ugW  = (const float*)d_in[10];  // [2][C][C]
  const float* ugb  = (const float*)d_in[11];
  const float* uaW  = (const float*)d_in[12];
  const float* uas  = (const float*)d_in[13];
  const float* uad  = (const float*)d_in[14];
  const float* uab  = (const float*)d_in[15];
  const float* lW   = (const float*)d_in[16];
  const float* lb   = (const float*)d_in[17];

  // bump allocator over d_ws
  char* base = (char*)d_ws;
  size_t off = 0;
  auto alloc = [&](size_t bytes) -> void* {
    size_t a = (off + 255) & ~(size_t)255;
    off = a + bytes;
    return (void*)(base + a);
  };
  auto fbuf = [&](size_t elems) { return (float*)alloc(elems * sizeof(float)); };
  auto ibuf = [&](size_t elems) { return (int*)alloc(elems * sizeof(int)); };

  // edge state per level/path
  int* mask0 = ibuf(E);
  int *cs1 = ibuf(E), *cd1 = ibuf(E), *cm1 = ibuf(E);
  int *as1 = ibuf(E), *ad1 = ibuf(E), *am1 = ibuf(E);
  int *cs2 = ibuf(E), *cd2 = ibuf(E), *cm2 = ibuf(E);
  int *as2 = ibuf(E), *ad2 = ibuf(E), *am2 = ibuf(E);
  // pooling scratch
  float* invn  = fbuf(1);
  float* score = fbuf(n0);
  int*   rank  = ibuf(n0);
  int*   pos   = ibuf(n0);
  int *perm_c1 = ibuf(k1), *perm_a1 = ibuf(k1);
  int *perm_c2 = ibuf(k2), *perm_a2 = ibuf(k2);
  // conv scratch
  float *deg = fbuf(n0), *dinv = fbuf(n0), *h = fbuf((size_t)n0 * C);
  float *al_s = fbuf(n0 * 4), *al_d = fbuf(n0 * 4);
  float *mx = fbuf(n0 * 4), *den = fbuf(n0 * 4);
  float* e_val = fbuf((size_t)(E + n0) * 4);
  // features
  float *xc0 = fbuf((size_t)n0 * C), *xa0 = fbuf((size_t)n0 * C);
  float *tmp = fbuf((size_t)n0 * C);
  float *xc1p = fbuf((size_t)n1 * C), *xa1p = fbuf((size_t)n1 * C);
  float *xc1 = fbuf((size_t)n1 * C), *xa1 = fbuf((size_t)n1 * C);
  float *xc2p = fbuf((size_t)n2 * C), *xa2p = fbuf((size_t)n2 * C);
  float *xc2 = fbuf((size_t)n2 * C), *xa2 = fbuf((size_t)n2 * C);
  float *xcu1 = fbuf((size_t)n1 * C), *xau1 = fbuf((size_t)n1 * C);
  float *xcn1 = fbuf((size_t)n1 * C), *xan1 = fbuf((size_t)n1 * C);
  float *xcu0 = fbuf((size_t)n0 * C), *xau0 = fbuf((size_t)n0 * C);
  float *xcn0 = fbuf((size_t)n0 * C), *xan0 = fbuf((size_t)n0 * C);
  (void)ws_size; (void)in_sizes; (void)n_in; (void)out_size;

  const size_t CC = (size_t)C * C;
  fill_i32_kernel<<<CDIV(E, 256), 256, 0, stream>>>(mask0, 1, E);

  // ---- down level 0 ----
  run_gcn(stream, x, src0, dst0, ew, mask0, dgW, dgb, xc0, n0, E, deg, dinv, h, true);
  add_kernel<<<CDIV(n0 * C, 256), 256, 0, stream>>>(x, xc0, tmp, n0 * C);
  run_gat(stream, tmp, src0, dst0, mask0, gaW, gas, gad, gab, xa0, n0, E,
          h, al_s, al_d, mx, den, e_val);

  // ---- pool to level 1 ----
  run_pool(stream, xa0, n0, k1, plw, invn, score, rank, pos, perm_a1, xa1p,
           src0, dst0, mask0, as1, ad1, am1, E);
  run_pool(stream, xc0, n0, k1, plw, invn, score, rank, pos, perm_c1, xc1p,
           src0, dst0, mask0, cs1, cd1, cm1, E);
  run_gcn(stream, xc1p, cs1, cd1, ew, cm1, dgW + CC, dgb + C, xc1, n1, E, deg, dinv, h, true);
  add_kernel<<<CDIV(n1 * C, 256), 256, 0, stream>>>(xc1, xa1p, tmp, n1 * C);
  run_gat(stream, tmp, as1, ad1, am1, gaW + CC, gas + 128, gad + 128, gab + C, xa1, n1, E,
          h, al_s, al_d, mx, den, e_val);

  // ---- pool to level 2 ----
  run_pool(stream, xa1, n1, k2, plw + C, invn, score, rank, pos, perm_a2, xa2p,
           as1, ad1, am1, as2, ad2, am2, E);
  run_pool(stream, xc1, n1, k2, plw + C, invn, score, rank, pos, perm_c2, xc2p,
           cs1, cd1, cm1, cs2, cd2, cm2, E);
  run_gcn(stream, xc2p, cs2, cd2, ew, cm2, dgW + 2 * CC, dgb + 2 * C, xc2, n2, E, deg, dinv, h, true);
  add_kernel<<<CDIV(n2 * C, 256), 256, 0, stream>>>(xc2, xa2p, tmp, n2 * C);
  run_gat(stream, tmp, as2, ad2, am2, gaW + 2 * CC, gas + 256, gad + 256, gab + 2 * C, xa2, n2, E,
          h, al_s, al_d, mx, den, e_val);

  // ---- up to level 1 ----
  copy_kernel<<<CDIV(n1 * C, 256), 256, 0, stream>>>(xc1, xcu1, n1 * C);
  unpool_scatter_kernel<<<CDIV(k2 * C, 256), 256, 0, stream>>>(perm_c2, xc2, xcu1, k2);
  copy_kernel<<<CDIV(n1 * C, 256), 256, 0, stream>>>(xa1, xau1, n1 * C);
  unpool_scatter_kernel<<<CDIV(k2 * C, 256), 256, 0, stream>>>(perm_a2, xa2, xau1, k2);
  run_gcn(stream, xcu1, cs1, cd1, ew, cm1, ugW, ugb, xcn1, n1, E, deg, dinv, h, true);
  add_kernel<<<CDIV(n1 * C, 256), 256, 0, stream>>>(xcn1, xau1, tmp, n1 * C);
  run_gat(stream, tmp, as1, ad1, am1, uaW, uas, uad, uab, xan1, n1, E,
          h, al_s, al_d, mx, den, e_val);

  // ---- up to level 0 ----
  copy_kernel<<<CDIV(n0 * C, 256), 256, 0, stream>>>(xc0, xcu0, n0 * C);
  unpool_scatter_kernel<<<CDIV(k1 * C, 256), 256, 0, stream>>>(perm_c1, xcn1, xcu0, k1);
  copy_kernel<<<CDIV(n0 * C, 256), 256, 0, stream>>>(xa0, xau0, n0 * C);
  unpool_scatter_kernel<<<CDIV(k1 * C, 256), 256, 0, stream>>>(perm_a1, xan1, xau0, k1);
  run_gcn(stream, xcu0, src0, dst0, ew, mask0, ugW + CC, ugb + C, xcn0, n0, E, deg, dinv, h, true);
  add_kernel<<<CDIV(n0 * C, 256), 256, 0, stream>>>(xcn0, xau0, tmp, n0 * C);
  run_gat(stream, tmp, src0, dst0, mask0, uaW + CC, uas + 128, uad + 128, uab + C, xan0, n0, E,
          h, al_s, al_d, mx, den, e_val);

  // ---- final GCN -> d_out ----
  add_kernel<<<CDIV(n0 * C, 256), 256, 0, stream>>>(xcn0, xan0, tmp, n0 * C);
  run_gcn(stream, tmp, src0, dst0, ew, mask0, lW, lb, (float*)d_out, n0, E,
          deg, dinv, h, false);
}